// pillarset_38534446580427
// MI455X (gfx1250) — compile-verified
//
#include <hip/hip_runtime.h>
#include <hip/hip_bf16.h>

// PointPillarsScatter for MI455X (gfx1250).
//
// Pure data movement: 327.7 MB of output stores dominate -> HBM-bound,
// ~15 us floor at 23.3 TB/s. Strategy:
//   1) winner map (d_ws): atomicMax(pillar index) per canvas cell ->
//      deterministic last-write-wins, resolves duplicates before any
//      feature data moves.
//   2) per-(b,y)-row workgroup builds the [64 ch][400 x] output tile
//      (100 KB) in LDS (feature gathers hit L2: 25.6 MB table is
//      L2-resident on a 192 MB L2), then a single TENSOR_STORE_FROM_LDS
//      DMA (2D descriptor, dim0 = 400 contiguous floats, dim1 = 64
//      channel rows strided by NY*NX) streams the tile to HBM.
//      Output stores are offloaded to the TDM engine; the wave only
//      pays LDS writes + an s_wait_tensorcnt.

#define NYD 400
#define NXD 400
#define CD  64
#define PLANE (NYD * NXD)   // 160000

typedef __attribute__((ext_vector_type(4))) unsigned int v4u;
typedef __attribute__((ext_vector_type(8))) int          v8i;
typedef __attribute__((ext_vector_type(4))) int          v4i;

// ---------------------------------------------------------------------------
// Kernel 1: winner[i] = -1 for all B*NY*NX cells (int4 stores, 5.1 MB).
// ---------------------------------------------------------------------------
__global__ void pp_init_winner(int4* __restrict__ w4, int n4) {
    int i = blockIdx.x * blockDim.x + threadIdx.x;
    if (i < n4) w4[i] = make_int4(-1, -1, -1, -1);
}

// ---------------------------------------------------------------------------
// Kernel 2: one thread per pillar. coords row = (b, z, y, x) as int4.
// atomicMax by pillar index == last-write-wins (torch advanced indexing).
// 100k atomics against an L2-resident 5.1 MB map.
// ---------------------------------------------------------------------------
__global__ void pp_scatter_winner(const int4* __restrict__ coords,
                                  int* __restrict__ winner, int P) {
    int p = blockIdx.x * blockDim.x + threadIdx.x;
    if (p >= P) return;
    int4 c = coords[p];                       // b, z, y, x
    int flat = c.x * PLANE + c.z * NXD + c.w; // b*NY*NX + y*NX + x
    atomicMax(&winner[flat], p);
}

// ---------------------------------------------------------------------------
// Kernel 3: one workgroup per (b, y) row. Build [CD][NXD] tile in LDS,
// then DMA it out with TENSOR_STORE_FROM_LDS.
// ---------------------------------------------------------------------------
__global__ void pp_gather_tdm(const float* __restrict__ feat,
                              const int* __restrict__ winner,
                              float* __restrict__ out) {
    extern __shared__ float tile[];           // [CD][NXD] = 102400 B
    const int tid = threadIdx.x;
    const int by  = blockIdx.x;               // b*NYD + y
    const int b   = by / NYD;
    const int y   = by - b * NYD;

    // Phase 1: zero the tile (6400 float4 DS stores across 256 threads).
    float4* t4 = (float4*)tile;
    const int n4 = (CD * NXD) / 4;            // 6400
    for (int i = tid; i < n4; i += blockDim.x)
        t4[i] = make_float4(0.f, 0.f, 0.f, 0.f);
    __syncthreads();

    // Phase 2: gather occupied cells of this row. Feature rows are 256 B
    // contiguous -> 16x b128 global gathers per occupied cell (L2 hits).
    // DS scatter tile[c*NXD + x]: consecutive lanes have consecutive x ->
    // consecutive LDS banks, conflict-free.
    const int rowbase = b * PLANE + y * NXD;
    for (int x = tid; x < NXD; x += blockDim.x) {
        int w = winner[rowbase + x];
        if (w >= 0) {
            const float4* f4 = (const float4*)(feat + (size_t)w * CD);
#pragma unroll
            for (int c4 = 0; c4 < CD / 4; ++c4) {
                float4 v = f4[c4];
                tile[(c4 * 4 + 0) * NXD + x] = v.x;
                tile[(c4 * 4 + 1) * NXD + x] = v.y;
                tile[(c4 * 4 + 2) * NXD + x] = v.z;
                tile[(c4 * 4 + 3) * NXD + x] = v.w;
            }
        }
    }
    __syncthreads();

    // Phase 3: wave 0 issues one TDM store for the whole tile.
    // 2D descriptor: dim0 = NXD contiguous floats, dim1 = CD rows at
    // stride PLANE elements (channel-major output planes).
    if (tid == 0) {
        unsigned long long gaddr =
            (unsigned long long)(uintptr_t)out +
            ((unsigned long long)b * CD * PLANE +
             (unsigned long long)y * NXD) * 4ull;
        unsigned lds_off = (unsigned)(uintptr_t)tile; // low 32 bits = LDS offset

        v4u g0;
        g0[0] = 1u;                                       // count=1, user mode
        g0[1] = lds_off;                                  // lds_addr (bytes)
        g0[2] = (unsigned)(gaddr & 0xFFFFFFFFu);          // global_addr[31:0]
        g0[3] = (unsigned)((gaddr >> 32) & 0x01FFFFFFu)   // global_addr[56:32]
                | 0x80000000u;                            // type=2 ("image")

        v8i g1;
        g1[0] = 0x00020000;                  // data_size=2 (4 B), mask=0
        g1[1] = (int)(((unsigned)NXD) << 16);// tensor_dim0[15:0] @ bits 63:48
        g1[2] = (int)(((unsigned)CD)  << 16);// tensor_dim1[15:0] @ bits 95:80
        g1[3] = (int)(((unsigned)NXD) << 16);// tile_dim0 @ bits 127:112
        g1[4] = CD;                          // tile_dim1 @ 143:128, tile_dim2=0
        g1[5] = PLANE;                       // tensor_dim0_stride[31:0]
        g1[6] = 0;                           // stride hi16, dim1_stride lo16
        g1[7] = 0;

        v4i gz4 = {0, 0, 0, 0};              // groups 2/3 unused (2-D tensor)
        v8i gz8 = {0, 0, 0, 0, 0, 0, 0, 0};  // extra group (clang-23 6-arg form)

        __builtin_amdgcn_tensor_store_from_lds(g0, g1, gz4, gz4, gz8, 0);
        __builtin_amdgcn_s_wait_tensorcnt(0); // keep LDS alive until DMA done
    }
}

// ---------------------------------------------------------------------------
extern "C" void kernel_launch(void* const* d_in, const int* in_sizes, int n_in,
                              void* d_out, int out_size, void* d_ws, size_t ws_size,
                              hipStream_t stream) {
    const float* feat   = (const float*)d_in[0];   // [P, 64] f32
    const int*   coords = (const int*)d_in[1];     // [P, 4]  i32
    (void)n_in; (void)ws_size;

    const int P = in_sizes[1] / 4;
    const int C = in_sizes[0] / P;                 // 64
    const int B = out_size / (C * PLANE);          // 8

    int* winner = (int*)d_ws;                      // B*NY*NX ints (5.12 MB)

    const int ncell  = B * PLANE;
    const int ncell4 = ncell / 4;
    pp_init_winner<<<(ncell4 + 255) / 256, 256, 0, stream>>>((int4*)winner, ncell4);

    pp_scatter_winner<<<(P + 255) / 256, 256, 0, stream>>>(
        (const int4*)coords, winner, P);

    pp_gather_tdm<<<B * NYD, 256, CD * NXD * sizeof(float), stream>>>(
        feat, winner, (float*)d_out);
}